// Sagittarius_48533130445523
// MI455X (gfx1250) — compile-verified
//
#include <hip/hip_runtime.h>
#include <hip/hip_bf16.h>
#include <math.h>

// Problem constants (fixed by the reference)
#define N_   8
#define TQ_  64
#define TK_  512
#define KQ_  128
#define H_   8
#define VD_  64
#define HKQ_ (H_ * KQ_)   // 1024
#define HVD_ (H_ * VD_)   // 512

typedef float v2f __attribute__((ext_vector_type(2)));
typedef float v8f __attribute__((ext_vector_type(8)));

// ---------------------------------------------------------------------------
// WMMA f32 16x16x4 tile GEMM: one wave (32 lanes) owns one 16x16 C tile.
// Fragment layouts per CDNA5 ISA 7.12.2:
//   A (16x4, MxK):  lanes 0-15 -> M=lane, {v0,v1}={K0,K1}; lanes 16-31 -> {K2,K3}
//   B (4x16, KxN):  lanes 0-15 -> N=lane, {v0,v1}={K0,K1}; lanes 16-31 -> {K2,K3}
//   C (16x16):      VGPR r: lanes 0-15 -> M=r, lanes 16-31 -> M=r+8, N=lane%16
// ---------------------------------------------------------------------------

// Projection: out_headmajor[n][h][t][d] = (A[row] @ W + bias)[h*KQ+d]
// A: (N*T, 128) row-major; W: (128, 1024); one wave per 16x16 tile.
__global__ void Sgt_proj_kernel(const float* __restrict__ A,
                                const float* __restrict__ W,
                                const float* __restrict__ bias,
                                float* __restrict__ out, int T) {
  const int lane = threadIdx.x;
  const int half = lane >> 4;       // 0: K0/K1, 1: K2/K3
  const int l    = lane & 15;
  const int tm   = blockIdx.x * 16; // row tile in flattened (n*T+t)
  const int tn   = blockIdx.y * 16; // col tile in [0,1024)

  v8f c = {};
  for (int k = 0; k < KQ_; k += 4) {
    const int ka = k + half * 2;
    v2f a, b;
    a.x = A[(size_t)(tm + l) * KQ_ + ka];
    a.y = A[(size_t)(tm + l) * KQ_ + ka + 1];
    b.x = W[(size_t)(ka)     * HKQ_ + tn + l];
    b.y = W[(size_t)(ka + 1) * HKQ_ + tn + l];
    c = __builtin_amdgcn_wmma_f32_16x16x4_f32(false, a, false, b,
                                              (short)0, c, false, false);
  }
  const int h  = tn >> 7;        // head of this 16-wide tile (16 | 128)
  const int d0 = tn & 127;
  const float bv = bias[tn + l];
  for (int r = 0; r < 8; ++r) {
    const int row = tm + r + half * 8;
    const int n = row / T, t = row % T;
    out[(((size_t)n * H_ + h) * T + t) * KQ_ + d0 + l] = c[r] + bv;
  }
}

// Scores: S[nh][q][k] = (hq[nh] @ hk[nh]^T)[q][k] / sqrt(128)
__global__ void Sgt_scores_kernel(const float* __restrict__ hq,
                                  const float* __restrict__ hk,
                                  float* __restrict__ scores) {
  const int lane = threadIdx.x;
  const int half = lane >> 4;
  const int l    = lane & 15;
  const int nh   = blockIdx.y;                       // n*H + h
  const int tm   = (blockIdx.x % (TQ_ / 16)) * 16;   // query tile
  const int tn   = (blockIdx.x / (TQ_ / 16)) * 16;   // key tile

  const float* Aq = hq + (size_t)nh * TQ_ * KQ_;
  const float* Bk = hk + (size_t)nh * TK_ * KQ_;

  v8f c = {};
  for (int k = 0; k < KQ_; k += 4) {
    const int ka = k + half * 2;
    v2f a, b;
    a.x = Aq[(size_t)(tm + l) * KQ_ + ka];
    a.y = Aq[(size_t)(tm + l) * KQ_ + ka + 1];
    // B = hk^T  =>  B[k][col] = hk[col][k]
    b.x = Bk[(size_t)(tn + l) * KQ_ + ka];
    b.y = Bk[(size_t)(tn + l) * KQ_ + ka + 1];
    c = __builtin_amdgcn_wmma_f32_16x16x4_f32(false, a, false, b,
                                              (short)0, c, false, false);
  }
  const float scale = 0.08838834764831845f; // 1/sqrt(128)
  for (int r = 0; r < 8; ++r) {
    const int q = tm + r + half * 8;
    scores[((size_t)nh * TQ_ + q) * TK_ + tn + l] = c[r] * scale;
  }
}

// Masked per-value-dim softmax over Tk + weighted sum with value.
// One block per (n,h,q). 256 threads = 4 k-groups x 64 value columns.
__global__ void Sgt_softmax_kernel(const float* __restrict__ scores,
                                   const int* __restrict__ mask,
                                   const float* __restrict__ value,
                                   float* __restrict__ p_attn,
                                   float* __restrict__ outh) {
  __shared__ float ss[TK_];
  __shared__ float red[4][VD_];

  const int b  = blockIdx.x;
  const int q  = b % TQ_;
  const int hh = (b / TQ_) % H_;
  const int n  = b / (TQ_ * H_);
  const int tid = threadIdx.x;

  const float* srow = scores + (((size_t)n * H_ + hh) * TQ_ + q) * TK_;
  ss[tid]       = srow[tid];
  ss[tid + 256] = srow[tid + 256];
  __syncthreads();

  const int v   = tid & (VD_ - 1);
  const int grp = tid >> 6;            // 0..3, each owns 128 k's
  const int k0  = grp * 128;

  const int*   mcol = mask  + (size_t)n * TK_ * VD_ + v;
  const float* vcol = value + (size_t)n * TK_ * VD_ + v;

  // Pass 1: online max/sum over this group's k range; cache mask bits.
  unsigned mb[4] = {0u, 0u, 0u, 0u};
  float m = -3.0e38f, s = 0.0f;
  for (int i = 0; i < 128; ++i) {
    const int k = k0 + i;
    const int mv = mcol[(size_t)k * VD_];
    if (mv) mb[i >> 5] |= (1u << (i & 31));
    const float x = mv ? ss[k] : -1.0e9f;
    const float mnew = fmaxf(m, x);
    s = s * expf(m - mnew) + expf(x - mnew);
    m = mnew;
  }

  // Cross-group reduction of (m, s).
  red[grp][v] = m;
  __syncthreads();
  const float M4 = fmaxf(fmaxf(red[0][v], red[1][v]),
                         fmaxf(red[2][v], red[3][v]));
  __syncthreads();
  red[grp][v] = s * expf(m - M4);
  __syncthreads();
  const float Z = red[0][v] + red[1][v] + red[2][v] + red[3][v];
  const float invZ = 1.0f / Z;
  __syncthreads();

  // Pass 2: emit p_attn (streamed, non-temporal) and accumulate out.
  float acc = 0.0f;
  float* prow = p_attn + ((((size_t)n * H_ + hh) * TQ_ + q) * TK_) * VD_ + v;
  for (int i = 0; i < 128; ++i) {
    const int k = k0 + i;
    const bool mv = (mb[i >> 5] >> (i & 31)) & 1u;
    const float x = mv ? ss[k] : -1.0e9f;
    const float p = expf(x - M4) * invZ;
    __builtin_nontemporal_store(p, prow + (size_t)k * VD_);
    acc += p * vcol[(size_t)k * VD_];
  }
  red[grp][v] = acc;
  __syncthreads();
  if (grp == 0) {
    const float o = red[0][v] + red[1][v] + red[2][v] + red[3][v];
    outh[(((size_t)n * H_ + hh) * TQ_ + q) * VD_ + v] = o;
  }
}

// Output projection: out_rep[(n,q)] = concat_h(outh[n][h][q]) @ Wo + bo
// Logical A (512 x 512): A[n*TQ+q][j] = outh[n][j>>6][q][j&63]
__global__ void Sgt_outproj_kernel(const float* __restrict__ outh,
                                   const float* __restrict__ Wo,
                                   const float* __restrict__ bo,
                                   float* __restrict__ out_rep) {
  const int lane = threadIdx.x;
  const int half = lane >> 4;
  const int l    = lane & 15;
  const int tm   = blockIdx.x * 16;   // rows of (n*TQ+q)
  const int tn   = blockIdx.y * 16;   // cols of VD

  const int row = tm + l;
  const int n = row >> 6, q = row & 63;

  v8f c = {};
  for (int k = 0; k < HVD_; k += 4) {
    const int kk = k + half * 2;
    v2f a, b;
    const int h0 = kk >> 6,       w0 = kk & 63;
    const int h1 = (kk + 1) >> 6, w1 = (kk + 1) & 63;
    a.x = outh[(((size_t)n * H_ + h0) * TQ_ + q) * VD_ + w0];
    a.y = outh[(((size_t)n * H_ + h1) * TQ_ + q) * VD_ + w1];
    b.x = Wo[(size_t)(kk)     * VD_ + tn + l];
    b.y = Wo[(size_t)(kk + 1) * VD_ + tn + l];
    c = __builtin_amdgcn_wmma_f32_16x16x4_f32(false, a, false, b,
                                              (short)0, c, false, false);
  }
  const float bv = bo[tn + l];
  for (int r = 0; r < 8; ++r) {
    out_rep[(size_t)(tm + r + half * 8) * VD_ + tn + l] = c[r] + bv;
  }
}

extern "C" void kernel_launch(void* const* d_in, const int* in_sizes, int n_in,
                              void* d_out, int out_size, void* d_ws, size_t ws_size,
                              hipStream_t stream) {
  (void)in_sizes; (void)n_in; (void)out_size; (void)ws_size;

  const float* value = (const float*)d_in[0];
  const float* key   = (const float*)d_in[1];
  const float* query = (const float*)d_in[2];
  const int*   mask  = (const int*)  d_in[3];
  const float* Wq    = (const float*)d_in[4];
  const float* bq    = (const float*)d_in[5];
  const float* Wk    = (const float*)d_in[6];
  const float* bk    = (const float*)d_in[7];
  const float* Wo    = (const float*)d_in[8];
  const float* bo    = (const float*)d_in[9];

  // Workspace layout (floats): hq | hk | scores | outh  (~27 MB total)
  float* ws      = (float*)d_ws;
  float* hq      = ws;                                       // 8*8*64*128
  float* hk      = hq + (size_t)N_ * H_ * TQ_ * KQ_;         // 8*8*512*128
  float* sc      = hk + (size_t)N_ * H_ * TK_ * KQ_;         // 8*8*64*512
  float* outh    = sc + (size_t)N_ * H_ * TQ_ * TK_;         // 8*8*64*64

  // Output layout: out_rep (N*TQ*VD) then p_attn (N*H*TQ*TK*VD)
  float* out_rep = (float*)d_out;
  float* p_attn  = out_rep + (size_t)N_ * TQ_ * VD_;

  // 1) Q projection: (N*TQ x 128) @ (128 x 1024) -> head-major hq
  dim3 gq(N_ * TQ_ / 16, HKQ_ / 16);       // (32, 64)
  Sgt_proj_kernel<<<gq, 32, 0, stream>>>(query, Wq, bq, hq, TQ_);

  // 2) K projection: (N*TK x 128) @ (128 x 1024) -> head-major hk
  dim3 gk(N_ * TK_ / 16, HKQ_ / 16);       // (256, 64)
  Sgt_proj_kernel<<<gk, 32, 0, stream>>>(key, Wk, bk, hk, TK_);

  // 3) Scores: per (n,h), (64x128)@(128x512) / sqrt(128)
  dim3 gs((TQ_ / 16) * (TK_ / 16), N_ * H_);  // (128, 64)
  Sgt_scores_kernel<<<gs, 32, 0, stream>>>(hq, hk, sc);

  // 4) Masked per-v softmax over Tk, stream p_attn, accumulate out
  Sgt_softmax_kernel<<<N_ * H_ * TQ_, 256, 0, stream>>>(sc, mask, value,
                                                        p_attn, outh);

  // 5) Output projection: (512 x 512) @ (512 x 64) + bo
  dim3 go(N_ * TQ_ / 16, VD_ / 16);        // (32, 4)
  Sgt_outproj_kernel<<<go, 32, 0, stream>>>(outh, Wo, bo, out_rep);
}